// PreImage_61211873902725
// MI455X (gfx1250) — compile-verified
//
#include <hip/hip_runtime.h>
#include <stdint.h>

// -----------------------------------------------------------------------------
// Graph scatter-add:  out[tgt[i]] += x[src[i]] * e[i]   (D_FEAT = 32 = wave32)
//
// lane = feature -> every x-row gather / out-row atomic is one coalesced
// 128B wave access. Edge metadata (src/tgt/e) streams global->LDS with CDNA5
// async-to-LDS ops (ASYNCcnt), double-buffered per wave; processing reads it
// back with uniform-address ds_loads (hardware broadcast, conflict-free).
// Accumulation: no-return global_atomic_add_f32 (STOREcnt, fire-and-forget).
// All hot addressing is 32-bit offsets off SGPR bases (GVS mode): x and out
// are 12.8 MB each, well within 32-bit reach.
// -----------------------------------------------------------------------------

#define D_FEAT          32
#define WAVES_PER_BLOCK 8
#define CHUNK           32            // edges staged per async step (1/lane)
#define CHUNKS_PER_WAVE 4             // 128 edges per wave
#define EDGES_PER_BLOCK (WAVES_PER_BLOCK * CHUNKS_PER_WAVE * CHUNK)  // 1024

// per-wave, per-buffer LDS layout (bytes):
//   [  0,256)  src  int64 x 32
//   [256,512)  tgt  int64 x 32
//   [512,640)  e    f32   x 32
#define BUF_BYTES 640
#define WAVE_LDS  (2 * BUF_BYTES)     // double buffered

__device__ __forceinline__ unsigned lds_offset(const void* p) {
  // low 32 bits of a generic pointer into LDS == byte offset within the
  // workgroup's LDS allocation (what async-to-LDS VDST expects)
  return (unsigned)(uintptr_t)p;
}

// saddr-form async gathers: SGPR 64-bit base + VGPR 32-bit byte offset
__device__ __forceinline__ void async_gather_b64(unsigned lds_off,
                                                 const void* base,
                                                 unsigned voff) {
  asm volatile("global_load_async_to_lds_b64 %0, %1, %2"
               :: "v"(lds_off), "v"(voff), "s"(base)
               : "memory");
}

__device__ __forceinline__ void async_gather_b32(unsigned lds_off,
                                                 const void* base,
                                                 unsigned voff) {
  asm volatile("global_load_async_to_lds_b32 %0, %1, %2"
               :: "v"(lds_off), "v"(voff), "s"(base)
               : "memory");
}

__global__ void __launch_bounds__(WAVES_PER_BLOCK * 32)
edge_scatter_kernel(const float* __restrict__ x,
                    const long long* __restrict__ src,
                    const long long* __restrict__ tgt,
                    const float* __restrict__ ew,
                    float* __restrict__ out,
                    int n_edges)
{
  __shared__ __align__(16) unsigned char smem[WAVES_PER_BLOCK * WAVE_LDS];

  const int      lane     = threadIdx.x & 31;       // feature index
  const int      wid      = threadIdx.x >> 5;
  const unsigned laneByte = (unsigned)lane << 2;    // lane*4 bytes
  unsigned char* wbase    = &smem[wid * WAVE_LDS];  // wave-private LDS slice

  const int waveId = blockIdx.x * WAVES_PER_BLOCK + wid;
  const int base0  = waveId * (CHUNKS_PER_WAVE * CHUNK);

  auto issue = [&](int chunk) {
    int base = base0 + chunk * CHUNK;
    int ei   = base + lane;
    if (ei >= n_edges) ei = n_edges - 1;            // clamp (masked at use)
    unsigned e8 = (unsigned)ei << 3;                // byte off into int64 array
    unsigned e4 = (unsigned)ei << 2;                // byte off into f32 array
    unsigned char* buf = wbase + (chunk & 1) * BUF_BYTES;
    async_gather_b64(lds_offset(buf +   0 + lane * 8), src, e8);
    async_gather_b64(lds_offset(buf + 256 + lane * 8), tgt, e8);
    async_gather_b32(lds_offset(buf + 512 + lane * 4), ew,  e4);
  };

  issue(0);                                         // prologue

  for (int c = 0; c < CHUNKS_PER_WAVE; ++c) {
    if (c + 1 < CHUNKS_PER_WAVE) {
      issue(c + 1);                                 // prefetch next chunk
      // 3 async ops still in flight (chunk c+1); chunk c is complete.
      asm volatile("s_wait_asynccnt %0" :: "i"(3) : "memory");
    } else {
      asm volatile("s_wait_asynccnt %0" :: "i"(0) : "memory");
    }

    const unsigned char* buf = wbase + (c & 1) * BUF_BYTES;
    int base = base0 + c * CHUNK;
    int cnt  = n_edges - base;
    if (cnt > CHUNK) cnt = CHUNK;

    #pragma unroll 4
    for (int j = 0; j < cnt; ++j) {
      // uniform LDS addresses -> broadcast ds_load, conflict-free
      unsigned s = (unsigned)*(const int*)  (buf +   0 + 8 * j); // low dword
      unsigned t = (unsigned)*(const int*)  (buf + 256 + 8 * j);
      float    w =           *(const float*)(buf + 512 + 4 * j);

      // 32-bit byte offsets off SGPR bases -> GVS addressing, no 64-bit VALU
      unsigned so = (s << 7) + laneByte;            // s * 128 + lane*4
      unsigned to = (t << 7) + laneByte;

      float xv = *(const float*)((const unsigned char*)x + so);  // 128B gather
      atomicAdd((float*)((unsigned char*)out + to), xv * w);     // no-rtn atomic
    }
  }
}

__global__ void zero_kernel(float* __restrict__ out, int n)
{
  int i4 = (blockIdx.x * blockDim.x + threadIdx.x) * 4;
  if (i4 + 3 < n) {
    *(float4*)(out + i4) = make_float4(0.f, 0.f, 0.f, 0.f);
  } else {
    for (int k = i4; k < n; ++k) out[k] = 0.f;
  }
}

extern "C" void kernel_launch(void* const* d_in, const int* in_sizes, int n_in,
                              void* d_out, int out_size, void* d_ws, size_t ws_size,
                              hipStream_t stream)
{
  (void)n_in; (void)d_ws; (void)ws_size;

  const float*     x = (const float*)d_in[0];
  const long long* a = (const long long*)d_in[1];   // int64 [2, n_edges]
  const float*     e = (const float*)d_in[2];
  float*           out = (float*)d_out;

  const int n_edges = in_sizes[2];
  const long long* src = a;                 // a[0, :]
  const long long* tgt = a + n_edges;       // a[1, :]

  // zero-init the output (harness poisons d_out; atomics need zeros)
  {
    int threads = 256;
    int vec4    = (out_size + 3) / 4;
    int blocks  = (vec4 + threads - 1) / threads;
    zero_kernel<<<blocks, threads, 0, stream>>>(out, out_size);
  }

  // scatter-add over edges
  {
    int blocks = (n_edges + EDGES_PER_BLOCK - 1) / EDGES_PER_BLOCK;
    edge_scatter_kernel<<<blocks, WAVES_PER_BLOCK * 32, 0, stream>>>(
        x, src, tgt, e, out, n_edges);
  }
}